// GraphBranch_82626580840515
// MI455X (gfx1250) — compile-verified
//
#include <hip/hip_runtime.h>
#include <math.h>

typedef __attribute__((ext_vector_type(2))) float v2f;
typedef __attribute__((ext_vector_type(8))) float v8f;

#define IN_FEATS 20
#define BATCH    16384
#define NNODES   (BATCH * IN_FEATS)   // 327680
#define HDIM     128
#define C1DIM    512
#define C2DIM    256
#define EPSV     1e-5f

__device__ __forceinline__ float elu1(float x) { return x > 0.f ? x : expm1f(x); }

// ---------------------------------------------------------------------------
// Layer 1: e1 = elu(h @ w1^T), h[n,f] = relu(x_n*wp_f + bp_f). K=128.
// Each wave owns TWO 16-wide output strips (o_base, o_base+128); all 64 B
// fragments live in registers, A fragments shared between the two strips.
// ---------------------------------------------------------------------------
__global__ __launch_bounds__(256)
void layer1_kernel(const float* __restrict__ x, const float* __restrict__ wp,
                   const float* __restrict__ bp, const float* __restrict__ w1,
                   float* __restrict__ e1, float* __restrict__ raw1)
{
    __shared__ float s_wp[HDIM], s_bp[HDIM];
    if (threadIdx.x < HDIM) { s_wp[threadIdx.x] = wp[threadIdx.x]; s_bp[threadIdx.x] = bp[threadIdx.x]; }
    __syncthreads();

    const int wave = threadIdx.x >> 5;
    const int lane = threadIdx.x & 31;
    const int col  = lane & 15;              // M row / N col within tile
    const int kk   = (lane >> 4) << 1;       // 0 or 2: K sub-offset
    const int o0   = blockIdx.y * 256 + wave * 16;   // blockIdx.y in [0,2)
    const int o1   = o0 + 128;

    v2f b0[32], b1[32];
    const float* w1r0 = w1 + (o0 + col) * HDIM + kk;
    const float* w1r1 = w1 + (o1 + col) * HDIM + kk;
#pragma unroll
    for (int s = 0; s < 32; ++s) {
        b0[s].x = w1r0[s * 4]; b0[s].y = w1r0[s * 4 + 1];
        b1[s].x = w1r1[s * 4]; b1[s].y = w1r1[s * 4 + 1];
    }

    for (int tile = blockIdx.x; tile < NNODES / 16; tile += gridDim.x) {
        const int m_base = tile * 16;
        const float xv = x[m_base + col];
        v8f c0 = {}, c1 = {};
#pragma unroll
        for (int s = 0; s < 32; ++s) {
            const int k = s * 4 + kk;
            v2f a;
            a.x = fmaxf(fmaf(xv, s_wp[k],     s_bp[k]),     0.f);
            a.y = fmaxf(fmaf(xv, s_wp[k + 1], s_bp[k + 1]), 0.f);
            c0 = __builtin_amdgcn_wmma_f32_16x16x4_f32(false, a, false, b0[s],
                                                       (short)0, c0, false, false);
            c1 = __builtin_amdgcn_wmma_f32_16x16x4_f32(false, a, false, b1[s],
                                                       (short)0, c1, false, false);
        }
        const int row_off = (lane >> 4) << 3; // lanes 16-31 hold M = 8..15
#pragma unroll
        for (int r = 0; r < 8; ++r) {
            const int row = m_base + row_off + r;
            const float v0 = c0[r], v1 = c1[r];
            if (row < IN_FEATS) {
                raw1[row * C1DIM + o0 + col] = v0;   // pre-mean rows for gat fixup
                raw1[row * C1DIM + o1 + col] = v1;
            }
            e1[(size_t)row * C1DIM + o0 + col] = elu1(v0);
            e1[(size_t)row * C1DIM + o1 + col] = elu1(v1);
        }
    }
}

// ---------------------------------------------------------------------------
// Layer 2: e2 = elu((a1*e1 + c1) @ w2^T). K=512. Each wave processes FOUR
// node tiles per iteration so each streamed w2 (B) fragment feeds 4 WMMAs,
// and the 4 accumulation chains are independent.
// ---------------------------------------------------------------------------
__global__ __launch_bounds__(256)
void layer2_kernel(const float* __restrict__ e1, const float* __restrict__ a1,
                   const float* __restrict__ c1, const float* __restrict__ w2,
                   float* __restrict__ e2, float* __restrict__ raw2)
{
    __shared__ float s_a1[C1DIM], s_c1[C1DIM];
    for (int i = threadIdx.x; i < C1DIM; i += blockDim.x) { s_a1[i] = a1[i]; s_c1[i] = c1[i]; }
    __syncthreads();

    const int wave = threadIdx.x >> 5;
    const int lane = threadIdx.x & 31;
    const int col  = lane & 15;
    const int kk   = (lane >> 4) << 1;
    const int o_base = blockIdx.y * 128 + wave * 16;    // blockIdx.y in [0,2)
    const float* w2row = w2 + (size_t)(o_base + col) * C1DIM + kk;

    for (int ts = blockIdx.x; ts < NNODES / 64; ts += gridDim.x) {
        const size_t m0 = (size_t)ts * 64;
        const float* e1r[4];
#pragma unroll
        for (int j = 0; j < 4; ++j)
            e1r[j] = e1 + (m0 + j * 16 + col) * C1DIM + kk;
        __builtin_prefetch(e1r[0] + (size_t)gridDim.x * 64 * C1DIM, 0, 1);

        v8f c[4] = {{}, {}, {}, {}};
#pragma unroll 4
        for (int s = 0; s < 128; ++s) {
            const int k = s * 4 + kk;
            v2f b; b.x = w2row[s * 4]; b.y = w2row[s * 4 + 1];
#pragma unroll
            for (int j = 0; j < 4; ++j) {
                v2f a;
                a.x = fmaf(s_a1[k],     e1r[j][s * 4],     s_c1[k]);
                a.y = fmaf(s_a1[k + 1], e1r[j][s * 4 + 1], s_c1[k + 1]);
                c[j] = __builtin_amdgcn_wmma_f32_16x16x4_f32(false, a, false, b,
                                                             (short)0, c[j], false, false);
            }
        }
        const int row_off = (lane >> 4) << 3;
#pragma unroll
        for (int j = 0; j < 4; ++j) {
#pragma unroll
            for (int r = 0; r < 8; ++r) {
                const size_t row = m0 + j * 16 + row_off + r;
                const float v = c[j][r];
                if (row < IN_FEATS) raw2[row * C2DIM + o_base + col] = v;
                e2[row * C2DIM + o_base + col] = elu1(v);
            }
        }
    }
}

// ---------------------------------------------------------------------------
// gat_mean fixup for node rows 0..19 (only rows touched by the edge index),
// then elu; overwrites e_out rows in place. Tiny: 1 block.
// ---------------------------------------------------------------------------
__global__ void fixup_kernel(const float* __restrict__ raw, const int* __restrict__ edge,
                             int nE, int C, float* __restrict__ e_out)
{
    const int* src = edge;
    const int* dst = edge + nE;
    for (int c = threadIdx.x; c < C; c += blockDim.x) {
        for (int i = 0; i < IN_FEATS; ++i) {
            float sum = 0.f; int cnt = 0;
            for (int e = 0; e < nE; ++e)
                if (dst[e] == i) { sum += raw[src[e] * C + c]; ++cnt; }
            const float v = (cnt > 0) ? (sum / (float)cnt) : raw[i * C + c];
            e_out[(size_t)i * C + c] = elu1(v);
        }
    }
}

// Per-channel sum / sum-of-squares over all N rows (coalesced, atomics on partials).
__global__ __launch_bounds__(256)
void stats_kernel(const float* __restrict__ e, int C, float* __restrict__ sums)
{
    const int nch = C / 256;                  // 2 for C1, 1 for C2
    float s[2] = {0.f, 0.f}, q[2] = {0.f, 0.f};
    for (int n = blockIdx.x; n < NNODES; n += gridDim.x) {
        const float* row = e + (size_t)n * C + threadIdx.x;
        for (int j = 0; j < nch; ++j) {
            const float v = row[j * 256];
            s[j] += v; q[j] += v * v;
        }
    }
    for (int j = 0; j < nch; ++j) {
        const int c = threadIdx.x + j * 256;
        atomicAdd(&sums[c], s[j]);
        atomicAdd(&sums[C + c], q[j]);
    }
}

// Fold batchnorm into per-channel affine: a = w*rsqrt(var+eps), cshift = b - mean*a.
__global__ void finalize_kernel(const float* __restrict__ sums, const float* __restrict__ bw,
                                const float* __restrict__ bb, int C,
                                float* __restrict__ a, float* __restrict__ cshift)
{
    const int c = blockIdx.x * blockDim.x + threadIdx.x;
    if (c < C) {
        const float m   = sums[c] / (float)NNODES;
        const float var = sums[C + c] / (float)NNODES - m * m;
        const float aa  = bw[c] * rsqrtf(var + EPSV);
        a[c] = aa;
        cshift[c] = bb[c] - m * aa;
    }
}

// Pool 20 nodes per batch row, apply bn2 affine, then LayerNorm over 256 channels.
__global__ __launch_bounds__(256)
void final_kernel(const float* __restrict__ e2, const float* __restrict__ a2,
                  const float* __restrict__ c2, const float* __restrict__ lw,
                  const float* __restrict__ lb, float* __restrict__ out)
{
    __shared__ float red[256];
    const int b = blockIdx.x, c = threadIdx.x;
    const float* base = e2 + (size_t)b * IN_FEATS * C2DIM + c;
    float s = 0.f;
#pragma unroll
    for (int i = 0; i < IN_FEATS; ++i) s += base[i * C2DIM];
    const float pooled = fmaf(a2[c], s * (1.f / IN_FEATS), c2[c]);

    red[c] = pooled; __syncthreads();
    for (int off = 128; off; off >>= 1) { if (c < off) red[c] += red[c + off]; __syncthreads(); }
    const float mu = red[0] * (1.f / 256.f); __syncthreads();
    const float d = pooled - mu;
    red[c] = d * d; __syncthreads();
    for (int off = 128; off; off >>= 1) { if (c < off) red[c] += red[c + off]; __syncthreads(); }
    const float var = red[0] * (1.f / 256.f);
    out[(size_t)b * C2DIM + c] = d * rsqrtf(var + EPSV) * lw[c] + lb[c];
}

// ---------------------------------------------------------------------------
extern "C" void kernel_launch(void* const* d_in, const int* in_sizes, int n_in,
                              void* d_out, int out_size, void* d_ws, size_t ws_size,
                              hipStream_t stream)
{
    const float* x     = (const float*)d_in[0];
    const float* wp    = (const float*)d_in[1];
    const float* bp    = (const float*)d_in[2];
    const float* w1    = (const float*)d_in[3];
    const float* w2    = (const float*)d_in[4];
    const float* bn1_w = (const float*)d_in[5];
    const float* bn1_b = (const float*)d_in[6];
    const float* bn2_w = (const float*)d_in[7];
    const float* bn2_b = (const float*)d_in[8];
    const float* ln_w  = (const float*)d_in[9];
    const float* ln_b  = (const float*)d_in[10];
    const int*   edge  = (const int*)d_in[11];
    const int nE = in_sizes[11] / 2;           // (2,140) row-major
    float* out = (float*)d_out;

    float* e1   = (float*)d_ws;                         // N*512
    float* e2   = e1 + (size_t)NNODES * C1DIM;          // N*256
    float* raw1 = e2 + (size_t)NNODES * C2DIM;          // 20*512
    float* raw2 = raw1 + IN_FEATS * C1DIM;              // 20*256
    float* st1  = raw2 + IN_FEATS * C2DIM;              // 2*512
    float* st2  = st1 + 2 * C1DIM;                      // 2*256
    float* a1   = st2 + 2 * C2DIM;
    float* c1s  = a1 + C1DIM;
    float* a2   = c1s + C1DIM;
    float* c2s  = a2 + C2DIM;
    (void)n_in; (void)out_size; (void)ws_size;

    hipMemsetAsync(st1, 0, (2 * C1DIM + 2 * C2DIM) * sizeof(float), stream);

    layer1_kernel<<<dim3(1280, 2), 256, 0, stream>>>(x, wp, bp, w1, e1, raw1);
    fixup_kernel<<<1, 256, 0, stream>>>(raw1, edge, nE, C1DIM, e1);
    stats_kernel<<<2048, 256, 0, stream>>>(e1, C1DIM, st1);
    finalize_kernel<<<2, 256, 0, stream>>>(st1, bn1_w, bn1_b, C1DIM, a1, c1s);

    layer2_kernel<<<dim3(640, 2), 256, 0, stream>>>(e1, a1, c1s, w2, e2, raw2);
    fixup_kernel<<<1, 256, 0, stream>>>(raw2, edge, nE, C2DIM, e2);
    stats_kernel<<<2048, 256, 0, stream>>>(e2, C2DIM, st2);
    finalize_kernel<<<1, 256, 0, stream>>>(st2, bn2_w, bn2_b, C2DIM, a2, c2s);

    final_kernel<<<BATCH, 256, 0, stream>>>(e2, a2, c2s, ln_w, ln_b, out);
}